// MixtralSparseMoeBlock_30657476559668
// MI455X (gfx1250) — compile-verified
//
#include <hip/hip_runtime.h>
#include <hip/hip_bf16.h>
#include <math.h>

#define NEXP 8
#define TTOK 8192
#define HDIM 1024
#define FDIM 4096

typedef float v2f __attribute__((ext_vector_type(2)));
typedef float v8f __attribute__((ext_vector_type(8)));

// ---------------------------------------------------------------------------
// Stage 1: router. One wave (32 lanes) per token. gate_w cached in LDS.
// Writes logits, and compacts (token, combine-weight) per expert into ws.
// ---------------------------------------------------------------------------
__global__ void moe_router_kernel(const float* __restrict__ x,
                                  const float* __restrict__ gate,
                                  float* __restrict__ logits_out,
                                  int* __restrict__ counts,
                                  int* __restrict__ tlist,
                                  float* __restrict__ wlist) {
  __shared__ float gl[NEXP * HDIM];          // 32 KB
  const int tid = threadIdx.x;
  for (int i = tid; i < NEXP * HDIM; i += 256) gl[i] = gate[i];
  __syncthreads();

  const int wave = tid >> 5;
  const int lane = tid & 31;
  const int t = blockIdx.x * 8 + wave;       // 8 tokens per block

  float acc[NEXP];
#pragma unroll
  for (int e = 0; e < NEXP; ++e) acc[e] = 0.0f;

  const float* xr = x + (size_t)t * HDIM;
  for (int h = lane; h < HDIM; h += 32) {
    const float xv = xr[h];
#pragma unroll
    for (int e = 0; e < NEXP; ++e) acc[e] += xv * gl[e * HDIM + h];
  }
#pragma unroll
  for (int e = 0; e < NEXP; ++e) {
#pragma unroll
    for (int off = 16; off > 0; off >>= 1)
      acc[e] += __shfl_xor(acc[e], off, 32);
  }

  if (lane == 0) {
    // write logits
#pragma unroll
    for (int e = 0; e < NEXP; ++e) logits_out[(size_t)t * NEXP + e] = acc[e];
    // softmax (fp32)
    float m = acc[0];
#pragma unroll
    for (int e = 1; e < NEXP; ++e) m = fmaxf(m, acc[e]);
    float p[NEXP], s = 0.0f;
#pragma unroll
    for (int e = 0; e < NEXP; ++e) { p[e] = __expf(acc[e] - m); s += p[e]; }
    const float inv = 1.0f / s;
#pragma unroll
    for (int e = 0; e < NEXP; ++e) p[e] *= inv;
    // top-2 (lowest index wins ties, matching lax.top_k)
    int e1 = 0;
#pragma unroll
    for (int e = 1; e < NEXP; ++e) if (p[e] > p[e1]) e1 = e;
    int e2 = (e1 == 0) ? 1 : 0;
#pragma unroll
    for (int e = 0; e < NEXP; ++e) if (e != e1 && p[e] > p[e2]) e2 = e;
    const float denom = p[e1] + p[e2];
    const float w1r = p[e1] / denom;
    const float w2r = p[e2] / denom;
    int i1 = atomicAdd(&counts[e1], 1);
    tlist[e1 * TTOK + i1] = t;  wlist[e1 * TTOK + i1] = w1r;
    int i2 = atomicAdd(&counts[e2], 1);
    tlist[e2 * TTOK + i2] = t;  wlist[e2 * TTOK + i2] = w2r;
  }
}

// ---------------------------------------------------------------------------
// Stage 2: fused SwiGLU expert FFN over compacted token lists.
// Block: 256 threads (8 waves), 32-token tile, full H=1024 output in regs.
//   GEMM1: g1/g3 = Xtile @ w1^T / w3^T  (M=32, N=64-chunk, K=1024)
//   h1 = silu(g1)*g3  -> LDS
//   GEMM2: out += h1 @ w2^T            (M=32, N=1024, K=64-chunk)
// All matmuls via v_wmma_f32_16x16x4_f32.
// ---------------------------------------------------------------------------
__global__ void moe_ffn_kernel(const float* __restrict__ x,
                               const float* __restrict__ w1,
                               const float* __restrict__ w3,
                               const float* __restrict__ w2,
                               float* __restrict__ out,
                               const int* __restrict__ counts,
                               const int* __restrict__ tlist,
                               const float* __restrict__ wlist) {
  constexpr int MB = 32;         // tokens per block
  constexpr int FN = 64;         // f-chunk
  constexpr int XS = HDIM + 2;   // padded X stride
  constexpr int HS = FN + 2;     // padded h1 stride

  __shared__ float Xs[MB * XS];  // ~128 KB
  __shared__ float H1s[MB * HS]; // ~8 KB
  __shared__ int   tok[MB];
  __shared__ float wt[MB];

  const int e    = blockIdx.x >> 8;     // 256 tiles per expert
  const int tile = blockIdx.x & 255;
  const int cnt  = counts[e];
  const int r0   = tile * MB;
  if (r0 >= cnt) return;                // uniform exit, EXEC stays full
  const int rowsValid = min(MB, cnt - r0);

  const int tid = threadIdx.x;
  if (tid < MB) {
    if (tid < rowsValid) {
      tok[tid] = tlist[e * TTOK + r0 + tid];
      wt[tid]  = wlist[e * TTOK + r0 + tid];
    } else { tok[tid] = 0; wt[tid] = 0.0f; }
  }
  __syncthreads();

  // gather x tile (invalid rows -> 0)
  for (int i = tid; i < MB * HDIM; i += 256) {
    const int r = i >> 10, c = i & (HDIM - 1);
    Xs[r * XS + c] = (r < rowsValid) ? x[(size_t)tok[r] * HDIM + c] : 0.0f;
  }
  __syncthreads();

  const int w    = tid >> 5;
  const int lane = tid & 31;
  const int m    = w & 1;          // M-tile (0..1)
  const int grp  = w >> 1;         // column group (0..3), 256 out cols each
  const int l15  = lane & 15;
  const int koff = (lane >> 4) * 2;
  const int arow = m * 16 + l15;

  const size_t eFH = (size_t)e * FDIM * HDIM;
  const size_t eHF = (size_t)e * HDIM * FDIM;

  v8f zero = {};
  v8f outacc[16];
#pragma unroll
  for (int j = 0; j < 16; ++j) outacc[j] = zero;

  for (int f0 = 0; f0 < FDIM; f0 += FN) {
    // ---- GEMM1: one 16x16 tile each of g1 and g3 per wave ----
    v8f g1 = zero, g3 = zero;
    const int fc = f0 + grp * 16 + l15;               // B fragment column
    const float* p1 = w1 + eFH + (size_t)fc * HDIM + koff;
    const float* p3 = w3 + eFH + (size_t)fc * HDIM + koff;
    const float* px = &Xs[arow * XS + koff];
    for (int k0 = 0; k0 < HDIM; k0 += 4) {
      v2f a  = *(const v2f*)(px + k0);
      v2f b1 = *(const v2f*)(p1 + k0);
      v2f b3 = *(const v2f*)(p3 + k0);
      g1 = __builtin_amdgcn_wmma_f32_16x16x4_f32(false, a, false, b1,
                                                 (short)0, g1, false, false);
      g3 = __builtin_amdgcn_wmma_f32_16x16x4_f32(false, a, false, b3,
                                                 (short)0, g3, false, false);
    }

    __syncthreads();   // prior GEMM2 readers of H1s are done
    {
      const int rr = m * 16 + (lane >> 4) * 8;
      const int cc = grp * 16 + l15;
#pragma unroll
      for (int vi = 0; vi < 8; ++vi) {
        const float a1 = g1[vi];
        const float a3 = g3[vi];
        const float sg = a1 / (1.0f + __expf(-a1));   // silu
        H1s[(rr + vi) * HS + cc] = sg * a3;
      }
    }
    __syncthreads();

    // ---- GEMM2: accumulate 16 output tiles per wave ----
#pragma unroll
    for (int j = 0; j < 16; ++j) {
      const int hcol = grp * 256 + j * 16 + l15;
      const float* p2 = w2 + eHF + (size_t)hcol * FDIM + f0 + koff;
      const float* ph = &H1s[arow * HS + koff];
      v8f accj = outacc[j];
      for (int k0 = 0; k0 < FN; k0 += 4) {
        v2f a = *(const v2f*)(ph + k0);
        v2f b = *(const v2f*)(p2 + k0);
        accj = __builtin_amdgcn_wmma_f32_16x16x4_f32(false, a, false, b,
                                                     (short)0, accj, false, false);
      }
      outacc[j] = accj;
    }
  }
  __syncthreads();

  // ---- scatter: out[token, h] += acc * routing_weight (2 adds/elem total) ----
#pragma unroll
  for (int j = 0; j < 16; ++j) {
    const int hcol = grp * 256 + j * 16 + l15;
#pragma unroll
    for (int vi = 0; vi < 8; ++vi) {
      const int r = m * 16 + (lane >> 4) * 8 + vi;
      if (r < rowsValid) {
        atomicAdd(&out[(size_t)tok[r] * HDIM + hcol], outacc[j][vi] * wt[r]);
      }
    }
  }
}

// ---------------------------------------------------------------------------
extern "C" void kernel_launch(void* const* d_in, const int* in_sizes, int n_in,
                              void* d_out, int out_size, void* d_ws, size_t ws_size,
                              hipStream_t stream) {
  const float* x    = (const float*)d_in[0];   // [B,S,H]
  const float* gate = (const float*)d_in[1];   // [E,H]
  const float* w1   = (const float*)d_in[2];   // [E,F,H]
  const float* w3   = (const float*)d_in[3];   // [E,F,H]
  const float* w2   = (const float*)d_in[4];   // [E,H,F]

  float* out    = (float*)d_out;                        // [T,H]
  float* logits = out + (size_t)TTOK * HDIM;            // [T,E]

  int*   counts = (int*)d_ws;
  int*   tlist  = (int*)((char*)d_ws + 256);
  float* wlist  = (float*)((char*)d_ws + 256 + (size_t)NEXP * TTOK * sizeof(int));

  hipMemsetAsync(out, 0, (size_t)TTOK * HDIM * sizeof(float), stream);
  hipMemsetAsync(counts, 0, NEXP * sizeof(int), stream);

  moe_router_kernel<<<TTOK / 8, 256, 0, stream>>>(x, gate, logits,
                                                  counts, tlist, wlist);
  moe_ffn_kernel<<<NEXP * (TTOK / 32), 256, 0, stream>>>(x, w1, w3, w2, out,
                                                         counts, tlist, wlist);
}